// GCNLayer_33397665693929
// MI455X (gfx1250) — compile-verified
//
#include <hip/hip_runtime.h>
#include <math.h>

typedef float v2f __attribute__((ext_vector_type(2)));
typedef float v8f __attribute__((ext_vector_type(8)));

#define D 256
#define MIN_NORM 1e-15f
#define HEPS 1e-7f
#define KC 64           // K-chunk staged in LDS per iteration
#define KCP 66          // padded LDS row stride (floats): keeps 8B align, spreads banks

// -------- block-wide sum reduction (blockDim.x == D == 256) --------
__device__ __forceinline__ float blockReduceSum(float v) {
    __shared__ float sbuf[D];
    int t = threadIdx.x;
    sbuf[t] = v;
    __syncthreads();
    for (int s = blockDim.x >> 1; s > 0; s >>= 1) {
        if (t < s) sbuf[t] += sbuf[t + s];
        __syncthreads();
    }
    float r = sbuf[0];
    __syncthreads();
    return r;
}

// -------- K1: logx = logmap0(x, c); one block (256 threads) per row --------
__global__ void k_logmap0(const float* __restrict__ x, const int* __restrict__ kp,
                          float* __restrict__ out) {
    int row = blockIdx.x, t = threadIdx.x;
    float c = (float)(*kp), Kc = 1.0f / c, sqrtK = sqrtf(Kc);
    float xv = x[(size_t)row * D + t];
    float sp = (t >= 1) ? xv : 0.0f;
    float ss = blockReduceSum(sp * sp);
    float ynorm = fmaxf(sqrtf(ss), MIN_NORM);
    float x0 = x[(size_t)row * D];
    float theta = fmaxf(x0 / sqrtK, 1.0f + HEPS);
    float coef = sqrtK * acoshf(theta) / ynorm;
    out[(size_t)row * D + t] = (t == 0) ? 0.0f : coef * xv;
}

// -------- pad W_gcn (255x255) into 256x256 with zero row0/col0 --------
__global__ void k_pad_w(const float* __restrict__ w, float* __restrict__ wp) {
    int idx = blockIdx.x * blockDim.x + threadIdx.x;
    int r = idx >> 8, cc = idx & 255;
    float v = 0.0f;
    if (r >= 1 && cc >= 1) v = w[(r - 1) * (D - 1) + (cc - 1)];
    wp[idx] = v;
}

// -------- f32 WMMA GEMM: C[M x ldn] = A[M x K] @ B[K x ldn] --------
// grid.x = M/16 row tiles, grid.y = ldn/16 col tiles, block = 32 (one wave/tile)
__global__ void k_gemm_wmma_f32(const float* __restrict__ A, const float* __restrict__ B,
                                float* __restrict__ Cm, int K, int ldn) {
    int lane = threadIdx.x;
    int ml = lane & 15;
    int kb = (lane >> 4) << 1;
    int row = blockIdx.x * 16 + ml;
    int col = blockIdx.y * 16 + ml;
    const float* Arow = A + (size_t)row * K;
    v8f acc = {0.f, 0.f, 0.f, 0.f, 0.f, 0.f, 0.f, 0.f};
    for (int k0 = 0; k0 < K; k0 += 4) {
        int k = k0 + kb;
        v2f a, b;
        a.x = Arow[k];
        a.y = Arow[k + 1];
        b.x = B[(size_t)k * ldn + col];
        b.y = B[(size_t)(k + 1) * ldn + col];
        acc = __builtin_amdgcn_wmma_f32_16x16x4_f32(false, a, false, b, (short)0, acc,
                                                    false, false);
    }
    int hh = lane >> 4;
#pragma unroll
    for (int r = 0; r < 8; ++r) {
        int orow = blockIdx.x * 16 + r + 8 * hh;
        Cm[(size_t)orow * ldn + col] = acc[r];
    }
}

// -------- K3: h_in = logmap0(proj(expmap0(tmp, c)), c) --------
__global__ void k_exp_proj_log(const float* __restrict__ tmp, const int* __restrict__ kp,
                               float* __restrict__ hin) {
    int row = blockIdx.x, t = threadIdx.x;
    float c = (float)(*kp), Kc = 1.0f / c, sqrtK = sqrtf(Kc);
    float uv = tmp[(size_t)row * D + t];
    float sp = (t >= 1) ? uv : 0.0f;
    float ss = blockReduceSum(sp * sp);
    float xn = fmaxf(sqrtf(ss), MIN_NORM);
    float theta = xn / sqrtK;
    float esp = (t >= 1) ? sqrtK * sinhf(theta) * uv / xn : 0.0f;
    float ss2 = blockReduceSum(esp * esp);
    float f0 = sqrtf(Kc + ss2);                     // proj recomputes x0
    float ynorm = fmaxf(sqrtf(ss2), MIN_NORM);
    float th2 = fmaxf(f0 / sqrtK, 1.0f + HEPS);
    float coef = sqrtK * acoshf(th2) / ynorm;
    hin[(size_t)row * D + t] = (t == 0) ? 0.0f : coef * esp;
}

// -------- g[j] = h[j,:] . a_att[D:2D] --------
__global__ void k_gvec(const float* __restrict__ h, const float* __restrict__ aatt,
                       float* __restrict__ g, int n) {
    int j = blockIdx.x * blockDim.x + threadIdx.x;
    if (j >= n) return;
    const float* a2 = aatt + D;
    const float* hr = h + (size_t)j * D;
    float s = 0.0f;
#pragma unroll 4
    for (int k = 0; k < D; ++k) s += hr[k] * a2[k];
    g[j] = s;
}

// -------- per-row online masked-softmax stats: mx[i], Z[i] (one adj pass) ---
__global__ void k_rowstats(const int* __restrict__ adj, const float* __restrict__ g,
                           float* __restrict__ mx, float* __restrict__ Zb, int n) {
    __shared__ float sm[D], ssv[D];
    int row = blockIdx.x, t = threadIdx.x;
    const int* ar = adj + (size_t)row * n;
    float m = -INFINITY, s = 0.0f;
    for (int j = t; j < n; j += blockDim.x) {
        if (ar[j] > 0) {
            float gv = g[j];
            if (gv > m) { s = s * __expf(m - gv) + 1.0f; m = gv; }
            else        { s += __expf(gv - m); }
        }
    }
    sm[t] = m; ssv[t] = s;
    __syncthreads();
    for (int st = blockDim.x >> 1; st > 0; st >>= 1) {
        if (t < st) {
            float m1 = sm[t], s1 = ssv[t], m2 = sm[t + st], s2 = ssv[t + st];
            float M = fmaxf(m1, m2);
            float comb = (M == -INFINITY) ? 0.0f
                       : (s1 * __expf(m1 - M) + s2 * __expf(m2 - M));
            sm[t] = M; ssv[t] = comb;
        }
        __syncthreads();
    }
    if (t == 0) { mx[row] = sm[0]; Zb[row] = ssv[0]; }
}

// -------- fused masked-softmax @ h, f32 WMMA, LDS-shared P tile -------------
// One 512-thread block (16 waves) per 16-row stripe. Wave w owns columns
// [16w, 16w+16). adj is streamed ONCE per stripe: the probability tile
// P[16][KC] is built in LDS by the whole block and consumed by all 16 waves.
__global__ void k_attn_wmma(const int* __restrict__ adj, const float* __restrict__ g,
                            const float* __restrict__ mx, const float* __restrict__ Zb,
                            const float* __restrict__ h, float* __restrict__ hp, int n) {
    __shared__ float P[16 * KCP];
    int tid = threadIdx.x;              // 0..511
    int lane = tid & 31;
    int wave = tid >> 5;                // 0..15 -> output column tile
    int ml = lane & 15;
    int kb = (lane >> 4) << 1;
    int hh = lane >> 4;
    int rowBase = blockIdx.x * 16;
    int col = wave * 16 + ml;

    v8f acc = {0.f, 0.f, 0.f, 0.f, 0.f, 0.f, 0.f, 0.f};

    for (int k0 = 0; k0 < n; k0 += KC) {
        __syncthreads();
        // stage P[16][KC]: 1024 elems, 512 threads -> 2 each
#pragma unroll
        for (int e = tid; e < 16 * KC; e += 512) {
            int i = e >> 6;             // KC == 64
            int kk = e & (KC - 1);
            int r = rowBase + i;
            const int* arow = adj + (size_t)r * n;
            __builtin_prefetch((const void*)(arow + k0 + kk + 4 * KC), 0, 1);
            int av = arow[k0 + kk];
            P[i * KCP + kk] = (av > 0) ? __expf(g[k0 + kk] - mx[r]) : 0.0f;
        }
        __syncthreads();
        // consume: each wave accumulates its 16x16 tile over this K-chunk
        for (int kk = 0; kk < KC; kk += 4) {
            int k = k0 + kk + kb;
            v2f a, b;
            a.x = P[ml * KCP + kk + kb];
            a.y = P[ml * KCP + kk + kb + 1];
            b.x = h[(size_t)k * D + col];
            b.y = h[(size_t)(k + 1) * D + col];
            acc = __builtin_amdgcn_wmma_f32_16x16x4_f32(false, a, false, b, (short)0,
                                                        acc, false, false);
        }
    }
#pragma unroll
    for (int r = 0; r < 8; ++r) {
        int orow = rowBase + r + 8 * hh;
        hp[(size_t)orow * D + col] = acc[r] / Zb[orow];
    }
}

// -------- final: elu -> proj -> logmap0 -> sigmoid -> expmap0(+proj) --------
__global__ void k_final(const float* __restrict__ hp, const int* __restrict__ kp,
                        float* __restrict__ out) {
    int row = blockIdx.x, t = threadIdx.x;
    float c = (float)(*kp), Kc = 1.0f / c, sqrtK = sqrtf(Kc);
    float v = hp[(size_t)row * D + t];
    float emb = (v > 0.0f) ? v : expm1f(v);         // elu
    float sp = (t >= 1) ? emb : 0.0f;
    float ss = blockReduceSum(sp * sp);
    float e0 = sqrtf(Kc + ss);                      // proj x0
    float ynorm = fmaxf(sqrtf(ss), MIN_NORM);
    float theta = fmaxf(e0 / sqrtK, 1.0f + HEPS);
    float coef = sqrtK * acoshf(theta) / ynorm;
    float lm = (t == 0) ? 0.0f : coef * sp;         // logmap0
    float sg = 1.0f / (1.0f + __expf(-lm));         // sigmoid
    float usp = (t >= 1) ? sg : 0.0f;
    float ss2 = blockReduceSum(usp * usp);
    float xn = fmaxf(sqrtf(ss2), MIN_NORM);
    float th2 = xn / sqrtK;
    float osp = (t >= 1) ? sqrtK * sinhf(th2) * usp / xn : 0.0f;
    float ss3 = blockReduceSum(osp * osp);          // proj inside expmap0
    out[(size_t)row * D + t] = (t == 0) ? sqrtf(Kc + ss3) : osp;
}

extern "C" void kernel_launch(void* const* d_in, const int* in_sizes, int n_in,
                              void* d_out, int out_size, void* d_ws, size_t ws_size,
                              hipStream_t stream) {
    const float* x    = (const float*)d_in[0];
    const int*   adj  = (const int*)d_in[1];
    const int*   kp   = (const int*)d_in[2];
    const float* Wgcn = (const float*)d_in[3];
    const float* Watt = (const float*)d_in[4];
    const float* aatt = (const float*)d_in[5];
    float* out = (float*)d_out;
    int n = in_sizes[0] / D;                        // 8192

    // workspace layout (floats)
    float* logx = (float*)d_ws;                     // n*D
    float* Wpad = logx + (size_t)n * D;             // D*D
    float* tmp  = Wpad + (size_t)D * D;             // n*D
    float* hin  = tmp  + (size_t)n * D;             // n*D
    float* h    = hin  + (size_t)n * D;             // n*D
    float* hp   = h    + (size_t)n * D;             // n*D
    float* g    = hp   + (size_t)n * D;             // n
    float* mx   = g + n;                            // n
    float* Zb   = mx + n;                           // n

    k_logmap0<<<n, D, 0, stream>>>(x, kp, logx);
    k_pad_w<<<D, D, 0, stream>>>(Wgcn, Wpad);
    k_gemm_wmma_f32<<<dim3(n / 16, D / 16), 32, 0, stream>>>(logx, Wpad, tmp, D, D);
    k_exp_proj_log<<<n, D, 0, stream>>>(tmp, kp, hin);
    k_gemm_wmma_f32<<<dim3(n / 16, D / 16), 32, 0, stream>>>(hin, Watt, h, D, D);
    k_gvec<<<(n + 255) / 256, 256, 0, stream>>>(h, aatt, g, n);
    k_rowstats<<<n, 256, 0, stream>>>(adj, g, mx, Zb, n);
    k_attn_wmma<<<n / 16, 512, 0, stream>>>(adj, g, mx, Zb, h, hp, n);
    k_final<<<n, D, 0, stream>>>(hp, kp, out);
}